// GraphPatchEmbed_4982162063639
// MI455X (gfx1250) — compile-verified
//
#include <hip/hip_runtime.h>
#include <hip/hip_bf16.h>

// ---------------------------------------------------------------------------
// GraphPatchEmbed, MI455X (gfx1250, wave32).
//
// Algebraic fusion: conv(p=2, stride=2) followed by the GCN linear is
//   xw = M @ patchvec,  M = gcn_w @ conv_w_flat  (192 x 12)
// so the 19.3 GFLOP 192x192 GEMM disappears. Remaining work is memory bound:
// ~214 MB ideal traffic -> ~9us at 23.3 TB/s. We fuse the patch projection
// and the grid-graph stencil in one kernel using an LDS tile with a 1-patch
// halo, and run the projection on V_WMMA_F32_16X16X4_F32.
//
// This revision removes all exec-masked conditional loads from the hot loops
// (clamped addresses + multiplicative 0/1 mask; conv_w zero-padded in LDS),
// so the WMMA K-loops are straight-line and loads use immediate offsets.
// ---------------------------------------------------------------------------

typedef __attribute__((ext_vector_type(2))) float v2f;
typedef __attribute__((ext_vector_type(4))) float v4f;
typedef __attribute__((ext_vector_type(8))) float v8f;

#define E_DIM   192
#define K_DIM   12          // C*p*p = 3*2*2
#define K_PAD   16
#define HH      256         // patch grid height (512/2)
#define WW      256
#define NN      (HH * WW)
#define TILE    16          // patches per block edge
#define HALO    18          // TILE + 2
#define NP      (HALO * HALO)    // 324 halo patches
#define CE      48          // feature chunk per block (192/48 = 4 chunks)
#define FT      (CE / 16)   // 3 feature wmma-tiles per chunk
#define NG      ((NP + 15) / 16) // 21 patch groups of 16
#define CH_STRIDE (512 * 512)    // x channel stride in floats

// ---------------------------------------------------------------------------
// Kernel 1: M_pad[192][16] = gcn_w[192x192] @ conv_w[192x12]  (k padded to 16)
// One workgroup, 12 waves, each wave owns one 16-row tile of M.
// conv_w is staged zero-padded into LDS so the 48-step K-loop has no branches.
// ---------------------------------------------------------------------------
__global__ __launch_bounds__(384) void build_m_kernel(
    const float* __restrict__ gcn_w,   // [192][192]
    const float* __restrict__ conv_w,  // [192][12] flat (e, c*4+u*2+v)
    float* __restrict__ m_pad)         // [192][16]
{
    __shared__ __align__(16) float cw_s[E_DIM * K_PAD];   // 12 KB, zero-padded

    for (int idx = threadIdx.x; idx < E_DIM * K_PAD; idx += 384) {
        const int row = idx >> 4, col = idx & 15;
        cw_s[idx] = (col < K_DIM) ? conv_w[row * K_DIM + col] : 0.f;
    }
    __syncthreads();

    const int lane = threadIdx.x & 31;
    const int wave = threadIdx.x >> 5;   // 0..11 -> e-row tile
    const int p16  = lane & 15;
    const int hi   = lane >> 4;

    const float* arow = gcn_w + (wave * 16 + p16) * E_DIM + 2 * hi;
    v8f acc = {};
    for (int s = 0; s < 48; ++s) {
        // A (16x4): lane holds gcn_w[erow][4s + 2*hi + {0,1}] (contig float2)
        v2f a = *(const v2f*)(arow + 4 * s);
        // B (4x16): rows = inner index e' = 4s + 2*hi + {0,1}, col = p16
        v2f b;
        b.x = cw_s[(4 * s + 2 * hi + 0) * K_PAD + p16];
        b.y = cw_s[(4 * s + 2 * hi + 1) * K_PAD + p16];
        acc = __builtin_amdgcn_wmma_f32_16x16x4_f32(
            false, a, false, b, (short)0, acc, false, false);
    }
    // D layout: VGPR r <-> row r + 8*hi, col = p16
#pragma unroll
    for (int r = 0; r < 8; ++r)
        m_pad[(wave * 16 + r + 8 * hi) * K_PAD + p16] = acc[r];
}

__device__ __forceinline__ float dinv_of(int i, int j) {
    // deg = 1 (self loop) + valid 4-neighbors; >= 1 always
    int deg = 1 + (i > 0) + (i < HH - 1) + (j > 0) + (j < WW - 1);
    return __frsqrt_rn((float)deg);
}

// ---------------------------------------------------------------------------
// Kernel 2 (fused): per block = (batch, 16x16 tile, feature chunk of 48).
// Phase 1: xw for tile+halo (324 patches x 48 feats) into LDS via WMMA.
// Phase 2: 5-point normalized stencil + bias, write out.
// ---------------------------------------------------------------------------
__global__ __launch_bounds__(256) void fused_gcn_kernel(
    const float* __restrict__ x,       // [4][3][512][512]
    const float* __restrict__ m_pad,   // [192][16]
    const float* __restrict__ gcn_b,   // [192]
    float* __restrict__ out)           // [4][65536][192]
{
    __shared__ __align__(16) float xw_s[NP * CE];   // 62,208 B

    const int tid  = threadIdx.x;
    const int lane = tid & 31;
    const int wave = tid >> 5;         // 8 waves
    const int p16  = lane & 15;
    const int hi   = lane >> 4;

    const int bi    = blockIdx.x >> 4;   // tile row
    const int bj    = blockIdx.x & 15;   // tile col
    const int e0    = blockIdx.y * CE;   // feature chunk base
    const int batch = blockIdx.z;

    const float* xb = x + (size_t)batch * 3 * CH_STRIDE;

    // ---- Phase 1: 21 patch-groups x 3 feature-tiles = 63 wave-tiles ----
    for (int t = wave; t < NG * FT; t += 8) {
        const int ft = t / NG;          // feature tile (0..2)
        const int g  = t % NG;          // patch group (0..20)
        const int q0 = g * 16 + p16;    // halo-linear patch id for this lane
        const int inq = (q0 < NP);
        const int q  = inq ? q0 : (NP - 1);   // clamp; masked below

        const int hr = q / HALO, hc = q % HALO;
        const int i  = bi * TILE + hr - 1;    // global patch coords (may be OOB)
        const int j  = bj * TILE + hc - 1;
        const float mask =
            (float)(inq & (i >= 0) & (i < HH) & (j >= 0) & (j < WW));
        const int ic = min(max(i, 0), HH - 1);
        const int jc = min(max(j, 0), WW - 1);

        // One base address; channel stride folds into the load offset.
        const float* xrow  = xb + (size_t)(2 * ic + hi) * 512 + 2 * jc;
        const float* abase = m_pad + (e0 + ft * 16 + p16) * K_PAD + 2 * hi;

        v8f acc = {};
#pragma unroll
        for (int s = 0; s < 3; ++s) {   // k = 4s..4s+3, channel c == s
            // A: M_pad[erow][4s + 2*hi + {0,1}]
            v2f a = *(const v2f*)(abase + 4 * s);
            // B: k = 4s+2*hi (v=0) and +1 (v=1) == x[b,s,2i+hi,2j..2j+1]
            v2f b = *(const v2f*)(xrow + (size_t)s * CH_STRIDE);
            b *= mask;                  // zero out-of-grid / tail lanes
            // full-EXEC wmma, straight-line loop body
            acc = __builtin_amdgcn_wmma_f32_16x16x4_f32(
                false, a, false, b, (short)0, acc, false, false);
        }
        if (inq) {
            float* dst = xw_s + (size_t)q0 * CE + ft * 16 + 8 * hi;
            *(v4f*)(dst + 0) = (v4f){acc[0], acc[1], acc[2], acc[3]};
            *(v4f*)(dst + 4) = (v4f){acc[4], acc[5], acc[6], acc[7]};
        }
    }
    __syncthreads();

    // ---- Phase 2: stencil. 256 patches x 12 float4-slots = 3072 slots ----
    const v4f* xv = (const v4f*)xw_s;   // stride 12 v4f per halo patch
    for (int slot = tid; slot < TILE * TILE * (CE / 4); slot += 256) {
        const int p  = slot / (CE / 4);
        const int f4 = slot % (CE / 4);
        const int ti = p >> 4, tj = p & 15;
        const int i  = bi * TILE + ti;
        const int j  = bj * TILE + tj;
        const int q  = (ti + 1) * HALO + (tj + 1);
        const int base = q * (CE / 4) + f4;

        const float sd = dinv_of(i, j);
        // halo xw of out-of-grid neighbors is exactly 0, so no branches needed
        v4f acc = sd * xv[base];
        acc += dinv_of(i - 1, j) * xv[base - HALO * (CE / 4)];
        acc += dinv_of(i + 1, j) * xv[base + HALO * (CE / 4)];
        acc += dinv_of(i, j - 1) * xv[base - (CE / 4)];
        acc += dinv_of(i, j + 1) * xv[base + (CE / 4)];

        v4f bias = *(const v4f*)(gcn_b + e0 + f4 * 4);
        v4f res  = acc * sd + bias;

        const size_t oidx = ((size_t)batch * NN + (size_t)(i * WW + j)) * E_DIM
                            + e0 + f4 * 4;
        *(v4f*)(out + oidx) = res;
    }
}

extern "C" void kernel_launch(void* const* d_in, const int* in_sizes, int n_in,
                              void* d_out, int out_size, void* d_ws, size_t ws_size,
                              hipStream_t stream) {
    const float* x      = (const float*)d_in[0];  // [4,3,512,512]
    const float* conv_w = (const float*)d_in[1];  // [192,3,2,2] == [192,12]
    const float* gcn_w  = (const float*)d_in[2];  // [192,192]
    const float* gcn_b  = (const float*)d_in[3];  // [192]
    float*       out    = (float*)d_out;          // [4,65536,192]
    float*       m_pad  = (float*)d_ws;           // 192*16 floats = 12 KB

    // Kernel 1: collapse gcn_w @ conv_w into M_pad (one workgroup, 12 waves)
    build_m_kernel<<<1, 384, 0, stream>>>(gcn_w, conv_w, m_pad);

    // Kernel 2: 256 tiles x 4 feature chunks x 4 batches
    fused_gcn_kernel<<<dim3(256, E_DIM / CE, 4), 256, 0, stream>>>(
        x, m_pad, gcn_b, out);
}